// OneToNLayer_2121713844698
// MI455X (gfx1250) — compile-verified
//
#include <hip/hip_runtime.h>

// OneToN sparse scatter layer for MI455X (gfx1250).
//   y[b, post[e]] += 100 * x[b, pre[e]],  e in [0, NNZ)
// Memory-bound (~65 MB mandatory HBM traffic, ~0.27 GFLOP) => LDS-resident
// scatter. CDNA5 paths used: global_load_async_to_lds_b32 (transposing DMA
// stage), ds_load_b128 gather, ds_add_f32 accumulation,
// global_store_async_from_lds_b128 writeback, s_wait_asynccnt.
// WMMA intentionally NOT used: densifying costs ~137 GFLOP for 0.27 GFLOP.

#define DIM_IN   8192
#define DIM_OUT  8192
#define BATCH    1024
#define N_FANOUT 16
#define NNZ      (DIM_IN * N_FANOUT)
#define WEIGHT_F 100.0f
#define ROWS     4      // batch rows per workgroup; LDS = 2*ROWS*32KB = 256KB
#define TPB      1024   // 32 waves (wave32)

#if defined(__AMDGCN__) && \
    __has_builtin(__builtin_amdgcn_global_load_async_to_lds_b32) && \
    __has_builtin(__builtin_amdgcn_s_wait_asynccnt)
#define USE_ASYNC_LD 1
#else
#define USE_ASYNC_LD 0
#endif

#if defined(__AMDGCN__) && \
    __has_builtin(__builtin_amdgcn_global_store_async_from_lds_b128) && \
    __has_builtin(__builtin_amdgcn_s_wait_asynccnt)
#define USE_ASYNC_ST 1
#else
#define USE_ASYNC_ST 0
#endif

#if defined(__AMDGCN__)
// Pointer types matching the builtin signatures (param0 = addrspace(1),
// param1 = addrspace(3); pointee type keyed to transfer width).
typedef int v4i __attribute__((ext_vector_type(4)));
typedef __attribute__((address_space(1))) v4i* g_v4i_p;  // global, b128
typedef __attribute__((address_space(3))) v4i* l_v4i_p;  // LDS,    b128
typedef __attribute__((address_space(1))) int* g_i32_p;  // global, b32
typedef __attribute__((address_space(3))) int* l_i32_p;  // LDS,    b32
#endif

__global__ __launch_bounds__(TPB) void one_to_n_scatter(
    const float* __restrict__ x,
    const int*   __restrict__ pre,
    const int*   __restrict__ post,
    float*       __restrict__ y)
{
    // xs4[c][r] = x[b0+r][c]  (row-interleaved so one ds_load_b128 fetches
    // column c for all 4 staged batch rows)
    __shared__ __align__(16) float xs4[DIM_IN][ROWS];   // 128 KB
    // ys[r][j]: flat layout == contiguous y[b0..b0+3][:] for DMA writeback
    __shared__ __align__(16) float ys[ROWS][DIM_OUT];   // 128 KB

    const int tid = threadIdx.x;
    const int b0  = blockIdx.x * ROWS;

    // ---- 1) zero the y accumulator tile (ds_store_b128) --------------------
    const float4 z4 = make_float4(0.f, 0.f, 0.f, 0.f);
    for (int q = tid; q < (ROWS * DIM_OUT) / 4; q += TPB)
        ((float4*)ys)[q] = z4;

    // ---- 2) transposing DMA stage: x[b0..b0+3][:] -> xs4[c][r] -------------
    {
        const float* gx = x + (size_t)b0 * DIM_IN;  // ROWS*DIM_IN contiguous
        for (int q = tid; q < ROWS * DIM_IN; q += TPB) {
            const int r = q >> 13;            // q / DIM_IN
            const int c = q & (DIM_IN - 1);   // q % DIM_IN
#if USE_ASYNC_LD
            __builtin_amdgcn_global_load_async_to_lds_b32(
                (g_i32_p)(gx + q), (l_i32_p)(&xs4[c][r]),
                /*offset=*/0, /*cpol=*/0);
#else
            xs4[c][r] = gx[q];
#endif
        }
#if USE_ASYNC_LD
        __builtin_amdgcn_s_wait_asynccnt(0);
#endif
    }
    __syncthreads();

    // ---- 3) edge scatter: b128 index loads, 1x ds_load_b128 gather,
    //         4x ds_add_f32 accumulate per edge ------------------------------
    {
        const int4* pre4  = (const int4*)pre;
        const int4* post4 = (const int4*)post;
        for (int q = tid; q < NNZ / 4; q += TPB) {
            const int4 c4 = pre4[q];
            const int4 j4 = post4[q];
#pragma unroll
            for (int u = 0; u < 4; ++u) {
                const int c = (u == 0) ? c4.x : (u == 1) ? c4.y : (u == 2) ? c4.z : c4.w;
                const int j = (u == 0) ? j4.x : (u == 1) ? j4.y : (u == 2) ? j4.z : j4.w;
                const float4 xv = *(const float4*)&xs4[c][0];   // ds_load_b128
                atomicAdd(&ys[0][j], WEIGHT_F * xv.x);
                atomicAdd(&ys[1][j], WEIGHT_F * xv.y);
                atomicAdd(&ys[2][j], WEIGHT_F * xv.z);
                atomicAdd(&ys[3][j], WEIGHT_F * xv.w);
            }
        }
    }
    __syncthreads();

    // ---- 4) writeback: ys tile is exactly y[b0..b0+3][:] -------------------
    {
        float* gy = y + (size_t)b0 * DIM_OUT;
        for (int q = tid; q < (ROWS * DIM_OUT) / 4; q += TPB) {
#if USE_ASYNC_ST
            __builtin_amdgcn_global_store_async_from_lds_b128(
                (g_v4i_p)(gy + 4 * q), (l_v4i_p)(&ys[0][0] + 4 * q),
                /*offset=*/0, /*cpol=*/0);
#else
            ((float4*)gy)[q] = ((const float4*)ys)[q];
#endif
        }
#if USE_ASYNC_ST
        __builtin_amdgcn_s_wait_asynccnt(0);   // drained before s_endpgm
#endif
    }
}

extern "C" void kernel_launch(void* const* d_in, const int* in_sizes, int n_in,
                              void* d_out, int out_size, void* d_ws, size_t ws_size,
                              hipStream_t stream)
{
    (void)in_sizes; (void)n_in; (void)out_size; (void)d_ws; (void)ws_size;
    const float* x    = (const float*)d_in[0];
    const int*   pre  = (const int*)d_in[1];
    const int*   post = (const int*)d_in[2];
    float*       y    = (float*)d_out;

    dim3 grid(BATCH / ROWS);   // 256 workgroups ~= one per WGP
    dim3 block(TPB);
    hipLaunchKernelGGL(one_to_n_scatter, grid, block, 0, stream, x, pre, post, y);
}